// GCNLayer_25314537242828
// MI455X (gfx1250) — compile-verified
//
#include <hip/hip_runtime.h>
#include <hip/hip_bf16.h>

#define N_NODES 100000
#define N_EDGES 1000000
#define DIM 64

typedef float v2f __attribute__((ext_vector_type(2)));
typedef float v8f __attribute__((ext_vector_type(8)));

// ---------------------------------------------------------------------------
// 1) zero out[], init deg[] = 1.0f (self loop)
// ---------------------------------------------------------------------------
__global__ void gcn_init_kernel(float* __restrict__ out, float* __restrict__ deg,
                                int out_elems, int n_nodes) {
    int gt = blockIdx.x * blockDim.x + threadIdx.x;
    if (gt < out_elems) out[gt] = 0.0f;
    if (gt < n_nodes)  deg[gt] = 1.0f;
}

// ---------------------------------------------------------------------------
// 2) deg[dst] += 1 per edge (hardware f32 atomics)
// ---------------------------------------------------------------------------
__global__ void gcn_degree_kernel(const long long* __restrict__ dst_idx,
                                  float* __restrict__ deg, int n_edges) {
    int e = blockIdx.x * blockDim.x + threadIdx.x;
    if (e >= n_edges) return;
    int d = (int)dst_idx[e];
    unsafeAtomicAdd(&deg[d], 1.0f);
}

// ---------------------------------------------------------------------------
// 3) dinv = rsqrt(deg) in place (deg >= 1 due to self-loops)
// ---------------------------------------------------------------------------
__global__ void gcn_rsqrt_kernel(float* __restrict__ deg, int n_nodes) {
    int i = blockIdx.x * blockDim.x + threadIdx.x;
    if (i >= n_nodes) return;
    float d = deg[i];
    deg[i] = (d > 0.0f) ? rsqrtf(d) : 0.0f;
}

// ---------------------------------------------------------------------------
// 4) h = x @ W^T via V_WMMA_F32_16X16X4_F32 (full fp32 precision)
//    block = 256 threads = 8 waves; wave (w) -> node tile (w>>2), out tile (w&3)
//    block covers 32 nodes x 64 out dims. 100000 % 32 == 0 -> no tail,
//    EXEC stays all-1s as WMMA requires.
//
//    A fragment (16x4 f32, 2 VGPRs): lane l holds A[l%16][k0 + 2*(l/16) + v]
//    B fragment (4x16 f32, 2 VGPRs): lane l holds B[k0 + 2*(l/16) + v][l%16]
//      with B[k][o] = W[o][k]  ->  both are contiguous float2 loads.
//    C/D (16x16 f32, 8 VGPRs):  acc[v] = D[v + 8*(l/16)][l%16]
// ---------------------------------------------------------------------------
__global__ void gcn_gemm_kernel(const float* __restrict__ x,
                                const float* __restrict__ W,
                                float* __restrict__ h) {
    const int wave = threadIdx.x >> 5;
    const int lane = threadIdx.x & 31;
    const int half = lane >> 4;      // 0 or 1
    const int lm   = lane & 15;

    const int node0 = blockIdx.x * 32 + (wave >> 2) * 16;
    const int out0  = (wave & 3) * 16;

    const float* xrow = x + (size_t)(node0 + lm) * DIM + half * 2;
    const float* wrow = W + (size_t)(out0  + lm) * DIM + half * 2;

    v8f acc = {};
#pragma unroll
    for (int k0 = 0; k0 < DIM; k0 += 4) {
        v2f a = *(const v2f*)(xrow + k0);
        v2f b = *(const v2f*)(wrow + k0);
        acc = __builtin_amdgcn_wmma_f32_16x16x4_f32(
            /*neg_a=*/false, a, /*neg_b=*/false, b,
            /*c_mod=*/(short)0, acc, /*reuse_a=*/false, /*reuse_b=*/false);
    }

#pragma unroll
    for (int v = 0; v < 8; ++v) {
        h[(size_t)(node0 + v + 8 * half) * DIM + out0 + lm] = acc[v];
    }
}

// ---------------------------------------------------------------------------
// 5) edge scatter: out[dst] += h[src] * dinv[src]*dinv[dst]
//    16 lanes per edge, 4 channels per lane (float4 gather, 4 f32 atomics)
// ---------------------------------------------------------------------------
__global__ void gcn_scatter_kernel(const long long* __restrict__ src_idx,
                                   const long long* __restrict__ dst_idx,
                                   const float* __restrict__ dinv,
                                   const float* __restrict__ h,
                                   float* __restrict__ out, int n_edges) {
    long long gt = (long long)blockIdx.x * blockDim.x + threadIdx.x;
    int e  = (int)(gt >> 4);
    int c4 = (int)(gt & 15) * 4;
    if (e >= n_edges) return;

    int s = (int)src_idx[e];
    int d = (int)dst_idx[e];
    float norm = dinv[s] * dinv[d];

    const float4 hv = *(const float4*)(h + (size_t)s * DIM + c4);
    float* o = out + (size_t)d * DIM + c4;
    unsafeAtomicAdd(o + 0, hv.x * norm);
    unsafeAtomicAdd(o + 1, hv.y * norm);
    unsafeAtomicAdd(o + 2, hv.z * norm);
    unsafeAtomicAdd(o + 3, hv.w * norm);
}

// ---------------------------------------------------------------------------
// 6) self-loop (norm = dinv[i]^2) + bias, non-atomic final pass
// ---------------------------------------------------------------------------
__global__ void gcn_selfloop_bias_kernel(const float* __restrict__ h,
                                         const float* __restrict__ dinv,
                                         const float* __restrict__ b,
                                         float* __restrict__ out, int n_nodes) {
    int gt = blockIdx.x * blockDim.x + threadIdx.x;
    int n = gt >> 6;
    int c = gt & 63;
    if (n >= n_nodes) return;
    float di = dinv[n];
    size_t idx = (size_t)n * DIM + c;
    out[idx] += h[idx] * di * di + b[c];
}

// ---------------------------------------------------------------------------
extern "C" void kernel_launch(void* const* d_in, const int* in_sizes, int n_in,
                              void* d_out, int out_size, void* d_ws, size_t ws_size,
                              hipStream_t stream) {
    const float*      x  = (const float*)d_in[0];                 // [N, 64]
    const long long*  ei = (const long long*)d_in[1];             // [2, E] int64
    const float*      W  = (const float*)d_in[2];                 // [64, 64]
    const float*      b  = (const float*)d_in[3];                 // [64]
    float*            out = (float*)d_out;                        // [N, 64]

    const long long* src = ei;            // row 0
    const long long* dst = ei + N_EDGES;  // row 1

    // workspace layout: dinv [N floats] | h [N*64 floats]
    float* dinv = (float*)d_ws;
    size_t h_off = ((size_t)N_NODES * sizeof(float) + 255) & ~(size_t)255;
    float* h = (float*)((char*)d_ws + h_off);

    const int out_elems = N_NODES * DIM;  // 6.4M

    // 1) init
    gcn_init_kernel<<<(out_elems + 255) / 256, 256, 0, stream>>>(out, dinv, out_elems, N_NODES);
    // 2) degrees
    gcn_degree_kernel<<<(N_EDGES + 255) / 256, 256, 0, stream>>>(dst, dinv, N_EDGES);
    // 3) rsqrt
    gcn_rsqrt_kernel<<<(N_NODES + 255) / 256, 256, 0, stream>>>(dinv, N_NODES);
    // 4) h = x @ W^T   (32 nodes per block; 100000/32 = 3125 exact)
    gcn_gemm_kernel<<<N_NODES / 32, 256, 0, stream>>>(x, W, h);
    // 5) edge scatter (16 lanes/edge)
    long long scat_threads = (long long)N_EDGES * 16;
    gcn_scatter_kernel<<<(int)((scat_threads + 255) / 256), 256, 0, stream>>>(
        src, dst, dinv, h, out, N_EDGES);
    // 6) self-loop + bias
    gcn_selfloop_bias_kernel<<<(out_elems + 255) / 256, 256, 0, stream>>>(h, dinv, b, out, N_NODES);
}